// GraphConv_2894807957807
// MI455X (gfx1250) — compile-verified
//
#include <hip/hip_runtime.h>

// GraphConv (DGCNN EdgeConv) for MI455X gfx1250.
//
// Key algebraic factorization:
//   y[b,o,n,k] = W1@(x_nbr - x_ctr) + W2@x_ctr
//              = (W1@x)[b,o,idx[b,n,k]] + ((W2-W1)@x)[b,o,n]
// so the big edge GEMM becomes two 64x64xN GEMMs + a gather.
//
// The dominant cost is the kNN Gram matrix inner = X^T X (4096x4096x64 per
// batch, fp32). We compute it with V_WMMA_F32_16X16X4_F32 fused with an
// in-register top-20 selection so the 537MB distance matrix never exists.

#define BDIM  8
#define CDIM  64
#define NDIM  4096
#define KNN   20
#define COUT  64

typedef float v2f __attribute__((ext_vector_type(2)));
typedef float v8f __attribute__((ext_vector_type(8)));

// ---------------------------------------------------------------- sq[b,n]
__global__ void __launch_bounds__(256)
sq_kernel(const float* __restrict__ x, float* __restrict__ sq) {
    int g = blockIdx.x * 256 + threadIdx.x;      // over B*N
    int b = g >> 12, n = g & (NDIM - 1);
    const float* X = x + (size_t)b * CDIM * NDIM + n;
    float s = 0.f;
    #pragma unroll 8
    for (int c = 0; c < CDIM; ++c) {
        float v = X[(size_t)c * NDIM];
        s += v * v;
    }
    sq[g] = s;
}

// ------------------------------------------------- fused WMMA kNN + top-20
// One wave owns 16 query columns (n-tile). B operand (queries) stays resident
// (16 K-steps x float2 = 32 VGPRs). Loop over 256 candidate m-tiles: load A
// fragments, 16x V_WMMA_F32_16X16X4_F32 (K=64), then per-lane top-20 insert.
// C layout: lane L holds column n0+(L&15), rows {r + 8*(L>>4)}; the two lane
// halves keep independent lists, merged via LDS at the end.
__global__ void __launch_bounds__(128)
knn_kernel(const float* __restrict__ x, const float* __restrict__ sq,
           int* __restrict__ idxout) {
    __shared__ float smV[128][KNN];
    __shared__ int   smI[128][KNN];

    const int lane = threadIdx.x & 31;
    const int wave = threadIdx.x >> 5;
    const int q    = blockIdx.x * 4 + wave;   // query tile id: 0 .. B*N/16-1
    const int b    = q >> 8;                  // / (N/16 = 256)
    const int n0   = (q & 255) << 4;
    const int col  = lane & 15;
    const int half = lane >> 4;

    const float* X   = x  + (size_t)b * CDIM * NDIM;
    const float* sqb = sq + (size_t)b * NDIM;

    // Resident query panel (B operand of D = A x B).
    // B[k][j] = x[c0+k][n0+j]; lane L, VGPR v: B[2*half+v][L&15].
    v2f Bf[16];
    #pragma unroll
    for (int t = 0; t < 16; ++t) {
        int c = 4 * t + 2 * half;
        Bf[t].x = X[(size_t)c       * NDIM + n0 + col];
        Bf[t].y = X[(size_t)(c + 1) * NDIM + n0 + col];
    }

    float lv[KNN]; int li[KNN];
    #pragma unroll
    for (int j = 0; j < KNN; ++j) { lv[j] = -3.0e38f; li[j] = 0; }

    for (int m0 = 0; m0 < NDIM; m0 += 16) {
        // Candidate panel (A operand): A[i][k] = x[c0+k][m0+i].
        v2f Af[16];
        #pragma unroll
        for (int t = 0; t < 16; ++t) {
            int c = 4 * t + 2 * half;
            Af[t].x = X[(size_t)c       * NDIM + m0 + col];
            Af[t].y = X[(size_t)(c + 1) * NDIM + m0 + col];
        }
        v8f acc = {};
        #pragma unroll
        for (int t = 0; t < 16; ++t)
            acc = __builtin_amdgcn_wmma_f32_16x16x4_f32(
                false, Af[t], false, Bf[t], (short)0, acc, false, false);

        // key = 2*inner - sq[m]  (the -sq[n] term is constant per query ->
        // does not change the ranking, so we drop it)
        const int mbase = m0 + 8 * half;
        #pragma unroll
        for (int r = 0; r < 8; ++r) {
            float key = 2.0f * acc[r] - sqb[mbase + r];
            if (key > lv[KNN - 1]) {
                float kv = key; int ki = mbase + r;
                #pragma unroll
                for (int j = 0; j < KNN; ++j) {
                    if (kv > lv[j]) {
                        float tv = lv[j]; lv[j] = kv; kv = tv;
                        int   ti = li[j]; li[j] = ki; ki = ti;
                    }
                }
            }
        }
    }

    #pragma unroll
    for (int j = 0; j < KNN; ++j) { smV[threadIdx.x][j] = lv[j]; smI[threadIdx.x][j] = li[j]; }
    __syncthreads();

    if (half == 0) {   // merge the two sorted half-lists for this query column
        const float* av = smV[threadIdx.x];      const int* ai = smI[threadIdx.x];
        const float* bv = smV[threadIdx.x + 16]; const int* bi = smI[threadIdx.x + 16];
        int* op = idxout + ((size_t)b * NDIM + n0 + col) * KNN;
        int ia = 0, ib = 0;
        for (int j = 0; j < KNN; ++j) {
            float a = av[ia], c2 = bv[ib];
            bool ta = (a > c2) || (a == c2 && ai[ia] < bi[ib]);
            op[j] = ta ? ai[ia] : bi[ib];
            if (ta) ++ia; else ++ib;
        }
    }
}

// ------------------------------------------ u = W1@x , v = (W2-W1)@x
__global__ void __launch_bounds__(256)
uv_kernel(const float* __restrict__ x, const float* __restrict__ W,
          float* __restrict__ u, float* __restrict__ v) {
    int g = blockIdx.x * 256 + threadIdx.x;      // over B*N
    int b = g >> 12, n = g & (NDIM - 1);
    const float* X = x + (size_t)b * CDIM * NDIM + n;
    float xc[CDIM];
    #pragma unroll
    for (int c = 0; c < CDIM; ++c) xc[c] = X[(size_t)c * NDIM];
    for (int o = 0; o < COUT; ++o) {
        const float* wr = W + o * 2 * CDIM;      // uniform -> scalar loads
        float au = 0.f, av = 0.f;
        #pragma unroll
        for (int c = 0; c < CDIM; ++c) {
            float w1 = wr[c], w2 = wr[CDIM + c];
            au += w1 * xc[c];
            av += (w2 - w1) * xc[c];
        }
        size_t off = ((size_t)b * COUT + o) * NDIM + n;
        u[off] = au; v[off] = av;
    }
}

// ------------------- y = gather(u) + v ; per-block partial BN statistics
__global__ void __launch_bounds__(256)
y_kernel(const float* __restrict__ u, const float* __restrict__ v,
         const int* __restrict__ idx, float* __restrict__ out,
         float* __restrict__ psum, float* __restrict__ psum2) {
    __shared__ float r1[256], r2[256];
    int n  = blockIdx.x * 256 + threadIdx.x;
    int bo = blockIdx.y;                         // b*64 + o
    int b  = bo >> 6;
    const float* ub = u + (size_t)bo * NDIM;
    float vv = v[(size_t)bo * NDIM + n];
    const int* ip = idx + ((size_t)b * NDIM + n) * KNN;
    float* op = out + ((size_t)bo * NDIM + n) * KNN;
    float s = 0.f, s2 = 0.f;
    #pragma unroll
    for (int k = 0; k < KNN; ++k) {
        float y = ub[ip[k]] + vv;
        op[k] = y;
        s += y; s2 += y * y;
    }
    r1[threadIdx.x] = s; r2[threadIdx.x] = s2;
    __syncthreads();
    for (int st = 128; st > 0; st >>= 1) {
        if (threadIdx.x < st) {
            r1[threadIdx.x] += r1[threadIdx.x + st];
            r2[threadIdx.x] += r2[threadIdx.x + st];
        }
        __syncthreads();
    }
    if (threadIdx.x == 0) {
        int pi = bo * gridDim.x + blockIdx.x;
        psum[pi] = r1[0]; psum2[pi] = r2[0];
    }
}

// ------------------------------- deterministic final reduce -> scale/shift
__global__ void stats_kernel(const float* __restrict__ psum, const float* __restrict__ psum2,
                             const float* __restrict__ gamma, const float* __restrict__ beta,
                             float* __restrict__ scale, float* __restrict__ shift) {
    int o = threadIdx.x;                         // 0..63
    float s = 0.f, s2 = 0.f;
    for (int b = 0; b < BDIM; ++b)
        for (int j = 0; j < NDIM / 256; ++j) {
            int pi = (b * COUT + o) * (NDIM / 256) + j;
            s += psum[pi]; s2 += psum2[pi];
        }
    const float M = (float)BDIM * NDIM * KNN;
    float mean = s / M;
    float var  = s2 / M - mean * mean;
    float rstd = rsqrtf(var + 1e-5f);
    float sc   = rstd * gamma[o];
    scale[o] = sc;
    shift[o] = beta[o] - mean * sc;
}

// --------------------------------------- in-place BN affine + leaky ReLU
__global__ void __launch_bounds__(256)
bn_kernel(float* __restrict__ out, const float* __restrict__ scale,
          const float* __restrict__ shift) {
    int i  = blockIdx.x * 256 + threadIdx.x;     // 0 .. N*KNN-1
    int bo = blockIdx.y;
    int o  = bo & 63;
    size_t off = (size_t)bo * (NDIM * KNN) + i;
    float y = out[off] * scale[o] + shift[o];
    out[off] = (y >= 0.f) ? y : 0.2f * y;
}

extern "C" void kernel_launch(void* const* d_in, const int* in_sizes, int n_in,
                              void* d_out, int out_size, void* d_ws, size_t ws_size,
                              hipStream_t stream) {
    const float* x     = (const float*)d_in[0];   // (8,64,4096)
    const float* W     = (const float*)d_in[1];   // (64,128)
    const float* gamma = (const float*)d_in[2];   // (64,)
    const float* beta  = (const float*)d_in[3];   // (64,)
    float* out = (float*)d_out;                   // (8,64,4096,20)

    // workspace layout (floats), ~19.6 MB total
    float* sqw   = (float*)d_ws;                          // B*N           = 32768
    int*   idxw  = (int*)(sqw + BDIM * NDIM);             // B*N*KNN       = 655360
    float* uw    = (float*)(idxw + BDIM * NDIM * KNN);    // B*COUT*N      = 2097152
    float* vw    = uw + (size_t)BDIM * COUT * NDIM;       // B*COUT*N      = 2097152
    float* psum  = vw + (size_t)BDIM * COUT * NDIM;       // 512*16        = 8192
    float* psum2 = psum + BDIM * COUT * (NDIM / 256);     // 8192
    float* scale = psum2 + BDIM * COUT * (NDIM / 256);    // 64
    float* shift = scale + COUT;                          // 64

    sq_kernel <<<(BDIM * NDIM) / 256, 256, 0, stream>>>(x, sqw);
    knn_kernel<<<(BDIM * NDIM / 16) / 4, 128, 0, stream>>>(x, sqw, idxw);
    uv_kernel <<<(BDIM * NDIM) / 256, 256, 0, stream>>>(x, W, uw, vw);
    y_kernel  <<<dim3(NDIM / 256, BDIM * COUT), 256, 0, stream>>>(uw, vw, idxw, out, psum, psum2);
    stats_kernel<<<1, COUT, 0, stream>>>(psum, psum2, gamma, beta, scale, shift);
    bn_kernel <<<dim3(NDIM * KNN / 256, BDIM * COUT), 256, 0, stream>>>(out, scale, shift);
}